// Voxelizer_13778255086206
// MI455X (gfx1250) — compile-verified
//
#include <hip/hip_runtime.h>
#include <hip/hip_bf16.h>

typedef unsigned int uint32;

#define RESV      256
#define CCH       4
#define HASH_BITS 19
#define HASH_SIZE (1u << HASH_BITS)       /* 524288 slots, load factor <= 0.5 */
#define HASH_MASK (HASH_SIZE - 1u)
#define RR        (RESV * RESV)

// CDNA5 native IEEE maximumNumber FP32 atomic (no-return form, device scope).
// ISA 15.18.3 opcode 82: GLOBAL_ATOMIC_MAX_NUM_F32  vaddr, vdata, off
__device__ __forceinline__ void atomic_max_num_f32(float* addr, float v) {
    asm volatile("global_atomic_max_num_f32 %0, %1, off scope:SCOPE_DEV"
                 :: "v"(addr), "v"(v)
                 : "memory");
}

// ---------------------------------------------------------------------------
// Kernel 1: initialize hash table, occupancy planes, and output (= -inf),
// using b128 stores (all regions are 16B-aligned, multiple-of-4 word counts).
// Word layout: [keys S][cnt S][sums 4S][occ 3RR][out n_out]
// ---------------------------------------------------------------------------
__global__ void vox_init(uint4* __restrict__ ws_vec, uint4* __restrict__ out_vec,
                         int n_out_vec) {
    size_t i = (size_t)blockIdx.x * blockDim.x + threadIdx.x;
    const size_t KEYS_V = HASH_SIZE / 4;                 // key vectors (= -1)
    const size_t WS_V   = (6ull * HASH_SIZE + 3ull * RR) / 4;  // total ws vectors
    if (i < KEYS_V) {
        ws_vec[i] = make_uint4(0xFFFFFFFFu, 0xFFFFFFFFu, 0xFFFFFFFFu, 0xFFFFFFFFu);
    } else if (i < WS_V) {
        ws_vec[i] = make_uint4(0u, 0u, 0u, 0u);          // cnt, sums, occ
    } else if (i < WS_V + (size_t)n_out_vec) {
        out_vec[i - WS_V] = make_uint4(0xFF800000u, 0xFF800000u,
                                       0xFF800000u, 0xFF800000u);  // -inf
    }
}

// ---------------------------------------------------------------------------
// Kernel 2: scatter points into the hash table (segment sums + counts),
// and count per-line occupancy exactly once per occupied voxel (CAS winner).
// All atomics are L2-resident (table ~12MB << 192MB L2).
// ---------------------------------------------------------------------------
__global__ void vox_scatter(const float4* __restrict__ coords,
                            const float4* __restrict__ feats,
                            int* __restrict__ keys, uint32* __restrict__ cnt,
                            float* __restrict__ sums, uint32* __restrict__ occ,
                            int n) {
    int i = blockIdx.x * blockDim.x + threadIdx.x;
    if (i >= n) return;

    float4 c = coords[i];   // b128 load
    int xi = min(max((int)floorf(c.x * (float)RESV), 0), RESV - 1);
    int yi = min(max((int)floorf(c.y * (float)RESV), 0), RESV - 1);
    int zi = min(max((int)floorf(c.z * (float)RESV), 0), RESV - 1);
    int b  = (int)c.w;      // B == 1 -> always 0, kept for fidelity
    int flat = (((b * RESV + xi) * RESV + yi) * RESV) + zi;

    uint32 slot = ((uint32)flat * 2654435761u) >> (32 - HASH_BITS);
    for (;;) {
        int prev = atomicCAS(&keys[slot], -1, flat);
        if (prev == -1) {
            // This thread claimed the voxel: record line occupancy once.
            atomicAdd(&occ[0 * RR + yi * RESV + zi], 1u);  // pH line (over x)
            atomicAdd(&occ[1 * RR + xi * RESV + zi], 1u);  // pW line (over y)
            atomicAdd(&occ[2 * RR + xi * RESV + yi], 1u);  // pZ line (over z)
            break;
        }
        if (prev == flat) break;
        slot = (slot + 1u) & HASH_MASK;
    }

    float4 f = feats[i];    // b128 load
    atomicAdd(&cnt[slot], 1u);
    atomicAdd(&sums[slot * 4 + 0], f.x);
    atomicAdd(&sums[slot * 4 + 1], f.y);
    atomicAdd(&sums[slot * 4 + 2], f.z);
    atomicAdd(&sums[slot * 4 + 3], f.w);
}

// ---------------------------------------------------------------------------
// Kernel 3: per occupied slot, compute mean (b128 load of the 4 channel sums)
// and fold it into the three projection planes with the CDNA5 fp32 max_num
// atomic. out[((p*C + c)*R + r)*R + s], p0:(r,s)=(y,z) p1:(x,z) p2:(x,y)
// ---------------------------------------------------------------------------
__global__ void vox_project(const int* __restrict__ keys,
                            const uint32* __restrict__ cnt,
                            const float4* __restrict__ sums,
                            float* __restrict__ out) {
    uint32 s = blockIdx.x * blockDim.x + threadIdx.x;
    if (s >= HASH_SIZE) return;
    int flat = keys[s];
    if (flat < 0) return;

    float inv = 1.0f / (float)cnt[s];
    int zi = flat & (RESV - 1);
    int yi = (flat >> 8) & (RESV - 1);
    int xi = (flat >> 16) & (RESV - 1);

    float4 sv = sums[s];    // b128 load of all 4 channel sums
    float m[CCH] = { sv.x * inv, sv.y * inv, sv.z * inv, sv.w * inv };

#pragma unroll
    for (int c = 0; c < CCH; ++c) {
        atomic_max_num_f32(&out[(0 * CCH + c) * RR + yi * RESV + zi], m[c]);
        atomic_max_num_f32(&out[(1 * CCH + c) * RR + xi * RESV + zi], m[c]);
        atomic_max_num_f32(&out[(2 * CCH + c) * RR + xi * RESV + yi], m[c]);
    }
}

// ---------------------------------------------------------------------------
// Kernel 4: empty voxels in a projection line contribute 0.0 to the max.
// Only a fully occupied 256-voxel line excludes the implicit 0.
// (-inf from init -> 0 for completely empty lines.)
// ---------------------------------------------------------------------------
__global__ void vox_fixup(float* __restrict__ out,
                          const uint32* __restrict__ occ, int n_out) {
    int i = blockIdx.x * blockDim.x + threadIdx.x;
    if (i >= n_out) return;
    int plane = i / (CCH * RR);   // 0..2
    int rs    = i % RR;           // r*R + s
    uint32 o  = occ[plane * RR + rs];
    float v   = out[i];
    out[i] = (o >= (uint32)RESV) ? v : fmaxf(v, 0.0f);
}

// ---------------------------------------------------------------------------
extern "C" void kernel_launch(void* const* d_in, const int* in_sizes, int n_in,
                              void* d_out, int out_size, void* d_ws, size_t ws_size,
                              hipStream_t stream) {
    const float* coords = (const float*)d_in[0];   // (N,4) f32
    const float* feats  = (const float*)d_in[1];   // (N,4) f32
    int n = in_sizes[0] / 4;                       // N points
    float* out = (float*)d_out;                    // 3*C*R*R f32

    // Workspace layout (~12.75 MB, fully L2-resident):
    char* ws = (char*)d_ws;
    int*    keys = (int*)ws;      ws += (size_t)HASH_SIZE * 4;
    uint32* cnt  = (uint32*)ws;   ws += (size_t)HASH_SIZE * 4;
    float*  sums = (float*)ws;    ws += (size_t)HASH_SIZE * 16;
    uint32* occ  = (uint32*)ws;   // 3*R*R u32

    const int BLK = 256;  // 8 wave32s
    size_t init_vec = (6ull * HASH_SIZE + 3ull * RR + (size_t)out_size) / 4;
    vox_init<<<(unsigned)((init_vec + BLK - 1) / BLK), BLK, 0, stream>>>(
        (uint4*)d_ws, (uint4*)d_out, out_size / 4);

    vox_scatter<<<(n + BLK - 1) / BLK, BLK, 0, stream>>>(
        (const float4*)coords, (const float4*)feats, keys, cnt, sums, occ, n);

    vox_project<<<(HASH_SIZE + BLK - 1) / BLK, BLK, 0, stream>>>(
        keys, cnt, (const float4*)sums, out);

    vox_fixup<<<(out_size + BLK - 1) / BLK, BLK, 0, stream>>>(
        out, occ, out_size);
}